// PointCloudNormals_20401094656332
// MI455X (gfx1250) — compile-verified
//
#include <hip/hip_runtime.h>
#include <hip/hip_bf16.h>
#include <float.h>

typedef __attribute__((ext_vector_type(2))) float v2f;
typedef __attribute__((ext_vector_type(8))) float v8f;

#define NPTS 8192
#define NCLOUD 2
#define KNN 16
#define EPSV 1e-12f

// ---------------------------------------------------------------------------
// Branchless sorted top-16 on packed u64 keys (hi 32 = float bits of d2,
// lo 32 = index). List kept descending: bk[0] = worst/max, bk[15] = best.
// Insert = single bubble pass (min/max chain), drops bk[0].
// ---------------------------------------------------------------------------
__device__ __forceinline__ void topk_insert_key(unsigned long long (&bk)[16],
                                                unsigned long long x) {
  if (x >= bk[0]) return;  // wave-level skip when no lane improves
  unsigned long long carry = x;
#pragma unroll
  for (int i = 15; i >= 0; --i) {
    unsigned long long bi = bk[i];
    bool c = carry < bi;
    bk[i] = c ? carry : bi;   // min
    carry = c ? bi : carry;   // max
  }
}

__device__ __forceinline__ unsigned long long pack_key(float d2, int idx) {
  unsigned u = __float_as_uint(fmaxf(d2, 0.0f));  // nonneg => bit-ordered
  return ((unsigned long long)u << 32) | (unsigned)idx;
}

// ---------------------------------------------------------------------------
// Smallest eigenvector of a symmetric 3x3 (Smith trigonometric eigenvalues,
// Cayley-Hamilton eigenvector: columns of (A-e1 I)(A-e2 I) span evec(e3)).
// ---------------------------------------------------------------------------
__device__ __forceinline__ void smallest_evec(float a00, float a01, float a02,
                                              float a11, float a12, float a22,
                                              float& ex, float& ey, float& ez) {
  const float third = 1.0f / 3.0f;
  float q = (a00 + a11 + a22) * third;
  float b00 = a00 - q, b11 = a11 - q, b22 = a22 - q;
  float p1 = a01 * a01 + a02 * a02 + a12 * a12;
  float p2 = b00 * b00 + b11 * b11 + b22 * b22 + 2.0f * p1;
  float p = sqrtf(fmaxf(p2 * (1.0f / 6.0f), 1e-30f));
  float invp = 1.0f / p;
  float det = b00 * (b11 * b22 - a12 * a12) - a01 * (a01 * b22 - a12 * a02) +
              a02 * (a01 * a12 - b11 * a02);
  float r = 0.5f * det * invp * invp * invp;
  r = fminf(fmaxf(r, -1.0f), 1.0f);
  float phi = acosf(r) * third;
  float e1 = q + 2.0f * p * cosf(phi);                        // largest
  float e3 = q + 2.0f * p * cosf(phi + 2.0943951023931953f);  // smallest
  float e2 = 3.0f * q - e1 - e3;

  float c1[9] = {a00 - e1, a01, a02, a01, a11 - e1, a12, a02, a12, a22 - e1};
  float c2[9] = {a00 - e2, a01, a02, a01, a11 - e2, a12, a02, a12, a22 - e2};
  float m[9];
#pragma unroll
  for (int i = 0; i < 3; ++i)
#pragma unroll
    for (int j = 0; j < 3; ++j) {
      float s = 0.0f;
#pragma unroll
      for (int k = 0; k < 3; ++k) s += c1[i * 3 + k] * c2[k * 3 + j];
      m[i * 3 + j] = s;
    }
  float n0 = m[0] * m[0] + m[3] * m[3] + m[6] * m[6];
  float n1 = m[1] * m[1] + m[4] * m[4] + m[7] * m[7];
  float n2 = m[2] * m[2] + m[5] * m[5] + m[8] * m[8];
  float vx, vy, vz, nn;
  if (n0 >= n1 && n0 >= n2) { vx = m[0]; vy = m[3]; vz = m[6]; nn = n0; }
  else if (n1 >= n2)        { vx = m[1]; vy = m[4]; vz = m[7]; nn = n1; }
  else                      { vx = m[2]; vy = m[5]; vz = m[8]; nn = n2; }
  float inv = rsqrtf(fmaxf(nn, 1e-30f));
  ex = vx * inv;
  ey = vy * inv;
  ez = vz * inv;
}

// ---------------------------------------------------------------------------
// Kernel 1: fused KNN (WMMA f32 16x16x4 distance tiles, 2 tiles/iter) +
// weighted covariance + smallest-eigenvector normal + sign disambiguation.
// Block = 128 threads = 4 waves; each wave owns 16 queries.
// ---------------------------------------------------------------------------
__global__ void __launch_bounds__(128)
knn_lrf_kernel(const float* __restrict__ verts, float* __restrict__ normals,
               int* __restrict__ knn_idx) {
  const int lane = threadIdx.x & 31;
  const int wave = threadIdx.x >> 5;
  const int cloud = blockIdx.y;
  const int qb = blockIdx.x * 64 + wave * 16;
  const float* V = verts + (size_t)cloud * NPTS * 3;

  const bool lower = lane < 16;
  const int qn = qb + (lane & 15);
  const float qx = V[qn * 3 + 0];
  const float qy = V[qn * 3 + 1];
  const float qz = V[qn * 3 + 2];
  const float qsq = qx * qx + qy * qy + qz * qz;

  // B (queries, 4x16): K0=x K1=y in lanes 0-15; K2=z K3=1 in lanes 16-31.
  v2f bmat;
  bmat.x = lower ? qx : qz;
  bmat.y = lower ? qy : 1.0f;

  unsigned long long bk[16];
  const unsigned long long init_key =
      ((unsigned long long)0x7F7FFFFFu << 32) | (unsigned)qn;  // FLT_MAX
#pragma unroll
  for (int i = 0; i < 16; ++i) bk[i] = init_key;

  const int mofs = lower ? 0 : 8;
  for (int t = 0; t < NPTS / 32; ++t) {
    const int base = t * 32;
    const int cn0 = base + (lane & 15);
    const int cn1 = cn0 + 16;

    float c0x = V[cn0 * 3 + 0], c0y = V[cn0 * 3 + 1], c0z = V[cn0 * 3 + 2];
    float c1x = V[cn1 * 3 + 0], c1y = V[cn1 * 3 + 1], c1z = V[cn1 * 3 + 2];
    float c0sq = c0x * c0x + c0y * c0y + c0z * c0z;
    float c1sq = c1x * c1x + c1y * c1y + c1z * c1z;

    // A (candidates, 16x4): K0=-2x K1=-2y lanes 0-15; K2=-2z K3=|c|^2 upper.
    v2f amat0, amat1;
    amat0.x = lower ? (-2.0f * c0x) : (-2.0f * c0z);
    amat0.y = lower ? (-2.0f * c0y) : c0sq;
    amat1.x = lower ? (-2.0f * c1x) : (-2.0f * c1z);
    amat1.y = lower ? (-2.0f * c1y) : c1sq;

    v8f zero = {0.f, 0.f, 0.f, 0.f, 0.f, 0.f, 0.f, 0.f};
    // D[M][N] = |c_M|^2 - 2 q_N . c_M  (EXEC is all-ones here)
    v8f acc0 = __builtin_amdgcn_wmma_f32_16x16x4_f32(false, amat0, false, bmat,
                                                     (short)0, zero, false,
                                                     false);
    v8f acc1 = __builtin_amdgcn_wmma_f32_16x16x4_f32(false, amat1, false, bmat,
                                                     (short)0, zero, false,
                                                     false);

    float d2a[8], d2b[8];
    float tmin = FLT_MAX;
#pragma unroll
    for (int rr = 0; rr < 8; ++rr) {
      d2a[rr] = acc0[rr] + qsq;
      d2b[rr] = acc1[rr] + qsq;
      tmin = fminf(tmin, fminf(d2a[rr], d2b[rr]));
    }
    const float worst = __uint_as_float((unsigned)(bk[0] >> 32));
    if (tmin < worst) {  // steady state: one s_cbranch_execz skips the tile
#pragma unroll
      for (int rr = 0; rr < 8; ++rr)
        topk_insert_key(bk, pack_key(d2a[rr], base + mofs + rr));
#pragma unroll
      for (int rr = 0; rr < 8; ++rr)
        topk_insert_key(bk, pack_key(d2b[rr], base + 16 + mofs + rr));
    }
  }

  // Merge upper-half-lane list into lower-half-lane list (union top-16).
#pragma unroll
  for (int j = 0; j < 16; ++j) {
    unsigned long long kv = __shfl(bk[j], lane | 16, 32);
    if (lower) topk_insert_key(bk, kv);
  }

  if (!lower) return;

  // --- weights + weighted covariance of neighbor offsets ---
  const float d2worst = __uint_as_float((unsigned)(bk[0] >> 32));
  const float radii = sqrtf(fmaxf(d2worst, EPSV));  // 16th-NN distance
  float nxv[16], nyv[16], nzv[16];
  float sw = 0.f;
  float c00 = 0.f, c01 = 0.f, c02 = 0.f, c11 = 0.f, c12 = 0.f, c22 = 0.f;
#pragma unroll
  for (int j = 0; j < 16; ++j) {
    int id = (int)(bk[j] & 0xFFFFFFFFull);
    float d2 = __uint_as_float((unsigned)(bk[j] >> 32));
    float nx = V[id * 3 + 0] - qx;
    float ny = V[id * 3 + 1] - qy;
    float nz = V[id * 3 + 2] - qz;
    nxv[j] = nx; nyv[j] = ny; nzv[j] = nz;
    float w = radii - sqrtf(fmaxf(d2, EPSV));
    sw += w;
    c00 += w * nx * nx; c01 += w * nx * ny; c02 += w * nx * nz;
    c11 += w * ny * ny; c12 += w * ny * nz; c22 += w * nz * nz;
  }
  float isw = 1.0f / sw;
  c00 *= isw; c01 *= isw; c02 *= isw; c11 *= isw; c12 *= isw; c22 *= isw;

  float zx, zy, zz;
  smallest_evec(c00, c01, c02, c11, c12, c22, zx, zy, zz);

  // Sign disambiguation: flip if fewer than K/2 neighbors agree.
  int cnt = 0;
#pragma unroll
  for (int j = 0; j < 16; ++j)
    cnt += (nxv[j] * zx + nyv[j] * zy + nzv[j] * zz >= 0.0f) ? 1 : 0;
  if (cnt < KNN / 2) { zx = -zx; zy = -zy; zz = -zz; }

  const size_t gp = (size_t)cloud * NPTS + qn;
  normals[gp * 3 + 0] = zx;
  normals[gp * 3 + 1] = zy;
  normals[gp * 3 + 2] = zz;
#pragma unroll
  for (int j = 0; j < 16; ++j)
    knn_idx[gp * 16 + j] = (int)(bk[j] & 0xFFFFFFFFull);
}

// ---------------------------------------------------------------------------
// Kernel 2: gather neighbor normals, center, 3x3 covariance / K, Frobenius-
// normalize 9-vector. One thread per point.
// ---------------------------------------------------------------------------
__global__ void __launch_bounds__(256)
normal_cov_kernel(const float* __restrict__ normals,
                  const int* __restrict__ knn_idx, float* __restrict__ out) {
  int p = blockIdx.x * blockDim.x + threadIdx.x;
  if (p >= NCLOUD * NPTS) return;
  int cloud = p >> 13;
  const float* nr = normals + (size_t)cloud * NPTS * 3;
  const int* id = knn_idx + (size_t)p * 16;

  float gx[16], gy[16], gz[16];
  float mx = 0.f, my = 0.f, mz = 0.f;
#pragma unroll
  for (int j = 0; j < 16; ++j) {
    int n = id[j];
    gx[j] = nr[n * 3 + 0];
    gy[j] = nr[n * 3 + 1];
    gz[j] = nr[n * 3 + 2];
    mx += gx[j]; my += gy[j]; mz += gz[j];
  }
  mx *= (1.0f / 16.0f); my *= (1.0f / 16.0f); mz *= (1.0f / 16.0f);

  float c00 = 0.f, c01 = 0.f, c02 = 0.f, c11 = 0.f, c12 = 0.f, c22 = 0.f;
#pragma unroll
  for (int j = 0; j < 16; ++j) {
    float dx = gx[j] - mx, dy = gy[j] - my, dz = gz[j] - mz;
    c00 += dx * dx; c01 += dx * dy; c02 += dx * dz;
    c11 += dy * dy; c12 += dy * dz; c22 += dz * dz;
  }
  const float s = 1.0f / 16.0f;
  float cv[9] = {c00 * s, c01 * s, c02 * s, c01 * s, c11 * s,
                 c12 * s, c02 * s, c12 * s, c22 * s};
  float nrm2 = 0.f;
#pragma unroll
  for (int j = 0; j < 9; ++j) nrm2 += cv[j] * cv[j];
  float inv = rsqrtf(fmaxf(nrm2, 1e-30f));
#pragma unroll
  for (int j = 0; j < 9; ++j) out[(size_t)p * 9 + j] = cv[j] * inv;
}

// ---------------------------------------------------------------------------
extern "C" void kernel_launch(void* const* d_in, const int* in_sizes, int n_in,
                              void* d_out, int out_size, void* d_ws,
                              size_t ws_size, hipStream_t stream) {
  const float* verts = (const float*)d_in[0];
  float* out = (float*)d_out;

  float* normals = (float*)d_ws;
  int* knn_idx =
      (int*)((char*)d_ws + (size_t)NCLOUD * NPTS * 3 * sizeof(float));

  dim3 g1(NPTS / 64, NCLOUD);
  knn_lrf_kernel<<<g1, dim3(128), 0, stream>>>(verts, normals, knn_idx);

  dim3 g2((NCLOUD * NPTS + 255) / 256);
  normal_cov_kernel<<<g2, dim3(256), 0, stream>>>(normals, knn_idx, out);
}